// VarianceAdaptor_17961553231993
// MI455X (gfx1250) — compile-verified
//
#include <hip/hip_runtime.h>
#include <hip/hip_bf16.h>

// VarianceAdaptor (FastSpeech2) for MI455X / gfx1250.
// Conv1d layers run as GEMMs on the native fp32 WMMA path
// (v_wmma_f32_16x16x4_f32) to keep full fp32 reference precision.
// B=32, L=512, H=F=256, K=3, NBINS=256, MAXLEN=2048.

#define B_      32
#define L_      512
#define H_      256
#define F_      256
#define MAXLEN_ 2048
#define NBINS_  256
#define EPS_    1e-5f

typedef __attribute__((ext_vector_type(2))) float v2f;
typedef __attribute__((ext_vector_type(8))) float v8f;

// ---------------------------------------------------------------------------
// Shared GEMM core: y_tile[16][256] = relu( conv3x1(xin)[16 rows] + bias )
// Block = 256 threads = 8 waves; each wave owns two 16x16 N-tiles.
// im2col: A[r, k=t*256+h] = xin[b, l0+r+t-1, h]  (zero pad at batch edges)
// ---------------------------------------------------------------------------
__device__ __forceinline__ void conv_gemm_relu(
    const float* __restrict__ xin, const float* __restrict__ w,
    const float* __restrict__ bias,
    float* __restrict__ x_tile,   // 18*256 floats (LDS)
    float* __restrict__ y_tile,   // 16*256 floats (LDS)
    int b, int l0)
{
  const int tid = threadIdx.x;

  // Stage 18 input rows (with conv halo) into LDS, zero-padded per batch.
  for (int i = tid; i < 18 * 256; i += 256) {
    int rr = i >> 8, cc = i & 255;
    int l = l0 + rr - 1;
    x_tile[i] = (l >= 0 && l < L_) ? xin[((b * L_ + l) << 8) + cc] : 0.f;
  }
  __syncthreads();

  const int wave = tid >> 5, lane = tid & 31;
  const int r    = lane & 15;            // A-frag row (M)
  const int koff = (lane >> 4) << 1;     // lanes 0-15: K 0,1 ; lanes 16-31: K 2,3
  const int nb   = wave << 5;            // 32 output cols per wave
  const int col  = lane & 15;            // B/C-frag column within tile

  v8f acc0 = {}; v8f acc1 = {};
  for (int k = 0; k < 3 * H_; k += 4) {
    const int t = k >> 8, h0 = k & 255;  // 4-chunks never straddle a tap
    const float* ap = &x_tile[((r + t) << 8) + h0 + koff];
    v2f a; a.x = ap[0]; a.y = ap[1];
    const float* wp = w + (size_t)(k + koff) * F_ + nb + col;
    v2f b0; b0.x = wp[0];  b0.y = wp[F_];
    v2f b1; b1.x = wp[16]; b1.y = wp[16 + F_];
    acc0 = __builtin_amdgcn_wmma_f32_16x16x4_f32(false, a, false, b0,
                                                 (short)0, acc0, false, false);
    acc1 = __builtin_amdgcn_wmma_f32_16x16x4_f32(false, a, false, b1,
                                                 (short)0, acc1, false, false);
  }

  // C/D layout: VGPR j, lanes 0-15 -> M=j ; lanes 16-31 -> M=j+8.
  const int rbase = (lane >> 4) << 3;
  for (int j = 0; j < 8; ++j) {
    int rr = j + rbase;
    int c0 = nb + col, c1 = c0 + 16;
    float v0 = acc0[j] + bias[c0]; v0 = v0 > 0.f ? v0 : 0.f;
    float v1 = acc1[j] + bias[c1]; v1 = v1 > 0.f ? v1 : 0.f;
    y_tile[(rr << 8) + c0] = v0;
    y_tile[(rr << 8) + c1] = v1;
  }
  __syncthreads();
}

// conv + relu + layernorm  ->  yout (B,L,F)
__global__ __launch_bounds__(256) void vp_conv_ln_kernel(
    const float* __restrict__ xin, const float* __restrict__ w,
    const float* __restrict__ bias, const float* __restrict__ g,
    const float* __restrict__ be, float* __restrict__ yout)
{
  __shared__ float x_tile[18 * 256];
  __shared__ float y_tile[16 * 256];
  const int b  = blockIdx.x >> 5;        // 32 row-blocks per batch
  const int l0 = (blockIdx.x & 31) << 4;
  conv_gemm_relu(xin, w, bias, x_tile, y_tile, b, l0);

  const int wave = threadIdx.x >> 5, lane = threadIdx.x & 31;
  for (int rr = wave * 2; rr < wave * 2 + 2; ++rr) {
    float s = 0.f, s2 = 0.f;
    for (int c = lane; c < 256; c += 32) {
      float v = y_tile[(rr << 8) + c];
      s += v; s2 += v * v;
    }
    for (int m = 16; m >= 1; m >>= 1) {
      s  += __shfl_xor(s,  m, 32);
      s2 += __shfl_xor(s2, m, 32);
    }
    float mu  = s  * (1.f / 256.f);
    float var = s2 * (1.f / 256.f) - mu * mu;
    float inv = rsqrtf(var + EPS_);
    float* out = yout + ((size_t)(b * L_ + l0 + rr) << 8);
    for (int c = lane; c < 256; c += 32) {
      float v = y_tile[(rr << 8) + c];
      out[c] = (v - mu) * inv * g[c] + be[c];
    }
  }
}

// conv + relu + layernorm + (F->1 linear head) + mask  ->  out (B*L)
__global__ __launch_bounds__(256) void vp_conv_ln_lin_kernel(
    const float* __restrict__ xin, const float* __restrict__ w,
    const float* __restrict__ bias, const float* __restrict__ g,
    const float* __restrict__ be, const float* __restrict__ lw,
    const float* __restrict__ lb, const int* __restrict__ mask,
    float* __restrict__ out)
{
  __shared__ float x_tile[18 * 256];
  __shared__ float y_tile[16 * 256];
  const int b  = blockIdx.x >> 5;
  const int l0 = (blockIdx.x & 31) << 4;
  conv_gemm_relu(xin, w, bias, x_tile, y_tile, b, l0);

  const int wave = threadIdx.x >> 5, lane = threadIdx.x & 31;
  for (int rr = wave * 2; rr < wave * 2 + 2; ++rr) {
    float s = 0.f, s2 = 0.f;
    for (int c = lane; c < 256; c += 32) {
      float v = y_tile[(rr << 8) + c];
      s += v; s2 += v * v;
    }
    for (int m = 16; m >= 1; m >>= 1) {
      s  += __shfl_xor(s,  m, 32);
      s2 += __shfl_xor(s2, m, 32);
    }
    float mu  = s  * (1.f / 256.f);
    float var = s2 * (1.f / 256.f) - mu * mu;
    float inv = rsqrtf(var + EPS_);
    float dot = 0.f;
    for (int c = lane; c < 256; c += 32) {
      float v  = y_tile[(rr << 8) + c];
      float hn = (v - mu) * inv * g[c] + be[c];
      dot += hn * lw[c];
    }
    for (int m = 16; m >= 1; m >>= 1) dot += __shfl_xor(dot, m, 32);
    if (lane == 0) {
      int row = b * L_ + l0 + rr;
      out[row] = mask[row] ? 0.f : (dot + lb[0]);
    }
  }
}

// x_out[b,l,:] = x_in[b,l,:] + emb[searchsorted_left(bins, tgt[b,l])]
__global__ __launch_bounds__(256) void add_emb_kernel(
    const float* __restrict__ xin, const float* __restrict__ tgt,
    const float* __restrict__ bins, const float* __restrict__ emb,
    float* __restrict__ xout)
{
  const int m = blockIdx.x;          // b*L + l
  const float v = tgt[m];
  int lo = 0, hi = NBINS_ - 1;       // 255 bin edges
  while (lo < hi) {
    int mid = (lo + hi) >> 1;
    if (bins[mid] < v) lo = mid + 1; else hi = mid;
  }
  const int c = threadIdx.x;
  xout[((size_t)m << 8) + c] = xin[((size_t)m << 8) + c] + emb[((size_t)lo << 8) + c];
}

// per-batch inclusive cumsum of durations; also emit duration_rounded + mel_len
__global__ __launch_bounds__(512) void cumsum_kernel(
    const int* __restrict__ dur, int* __restrict__ cum,
    float* __restrict__ out_dur, float* __restrict__ out_mellen)
{
  __shared__ int s[L_];
  const int b = blockIdx.x, t = threadIdx.x;
  const int d = dur[b * L_ + t];
  s[t] = d;
  __syncthreads();
  for (int off = 1; off < L_; off <<= 1) {
    int add = (t >= off) ? s[t - off] : 0;
    __syncthreads();
    s[t] += add;
    __syncthreads();
  }
  cum[b * L_ + t]     = s[t];
  out_dur[b * L_ + t] = (float)d;
  if (t == L_ - 1) out_mellen[b] = (float)s[t];
}

// length regulator: gather source frames by searchsorted_right(cum, frame)
__global__ __launch_bounds__(256) void lenreg_kernel(
    const float* __restrict__ xin, const int* __restrict__ cum,
    float* __restrict__ xout, float* __restrict__ melmask)
{
  const int b = blockIdx.x >> 11;          // MAXLEN = 2048
  const int f = blockIdx.x & (MAXLEN_ - 1);
  const int* c = cum + b * L_;
  const int mel_len = c[L_ - 1];
  const bool valid = f < mel_len;
  int lo = 0, hi = L_;
  while (lo < hi) {
    int mid = (lo + hi) >> 1;
    if (c[mid] <= f) lo = mid + 1; else hi = mid;
  }
  const int idx = lo < (L_ - 1) ? lo : (L_ - 1);
  const int t = threadIdx.x;
  xout[(((size_t)b << 11) + f) * H_ + t] =
      valid ? xin[((size_t)(b * L_ + idx) << 8) + t] : 0.f;
  if (t == 0) melmask[((size_t)b << 11) + f] = valid ? 0.f : 1.f;
}

// ---------------------------------------------------------------------------
extern "C" void kernel_launch(void* const* d_in, const int* in_sizes, int n_in,
                              void* d_out, int out_size, void* d_ws, size_t ws_size,
                              hipStream_t stream) {
  (void)in_sizes; (void)n_in; (void)out_size; (void)ws_size;

  const float* x        = (const float*)d_in[0];
  const int*   src_mask = (const int*)  d_in[1];
  const float* pitch_t  = (const float*)d_in[2];
  const float* energy_t = (const float*)d_in[3];
  const int*   dur_t    = (const int*)  d_in[4];
  // d_in[5] = max_len scalar (fixed MAXLEN_)
  // params pytree-flattened (alphabetical): dur{b1,b2,be1,be2,g1,g2,lb,lw,w1,w2},
  // energy{...}, energy_emb, pitch{...}, pitch_emb, then pitch_bins, energy_bins.
  auto P = [&](int i) { return (const float*)d_in[i]; };
  const float *d_b1=P(6),  *d_b2=P(7),  *d_be1=P(8),  *d_be2=P(9),  *d_g1=P(10),
              *d_g2=P(11), *d_lb=P(12), *d_lw=P(13),  *d_w1=P(14),  *d_w2=P(15);
  const float *e_b1=P(16), *e_b2=P(17), *e_be1=P(18), *e_be2=P(19), *e_g1=P(20),
              *e_g2=P(21), *e_lb=P(22), *e_lw=P(23),  *e_w1=P(24),  *e_w2=P(25);
  const float *energy_emb = P(26);
  const float *p_b1=P(27), *p_b2=P(28), *p_be1=P(29), *p_be2=P(30), *p_g1=P(31),
              *p_g2=P(32), *p_lb=P(33), *p_lw=P(34),  *p_w1=P(35),  *p_w2=P(36);
  const float *pitch_emb   = P(37);
  const float *pitch_bins  = P(38);
  const float *energy_bins = P(39);

  float* out = (float*)d_out;
  const size_t o_x       = 0;
  const size_t o_pitch   = (size_t)B_ * MAXLEN_ * H_;
  const size_t o_energy  = o_pitch  + (size_t)B_ * L_;
  const size_t o_logdur  = o_energy + (size_t)B_ * L_;
  const size_t o_dur     = o_logdur + (size_t)B_ * L_;
  const size_t o_mellen  = o_dur    + (size_t)B_ * L_;
  const size_t o_melmask = o_mellen + (size_t)B_;

  float* buf0 = (float*)d_ws;                         // (B,L,F) conv1 output
  float* buf1 = buf0 + (size_t)B_ * L_ * F_;          // (B,L,H) x2 / x3
  int*   cum  = (int*)(buf1 + (size_t)B_ * L_ * H_);  // (B,L) cumsum

  const dim3 blk(256), gconv(B_ * L_ / 16), grow(B_ * L_), glr(B_ * MAXLEN_);

  // duration predictor (on x)
  vp_conv_ln_kernel    <<<gconv, blk, 0, stream>>>(x, d_w1, d_b1, d_g1, d_be1, buf0);
  vp_conv_ln_lin_kernel<<<gconv, blk, 0, stream>>>(buf0, d_w2, d_b2, d_g2, d_be2,
                                                   d_lw, d_lb, src_mask, out + o_logdur);
  // pitch predictor (on x)
  vp_conv_ln_kernel    <<<gconv, blk, 0, stream>>>(x, p_w1, p_b1, p_g1, p_be1, buf0);
  vp_conv_ln_lin_kernel<<<gconv, blk, 0, stream>>>(buf0, p_w2, p_b2, p_g2, p_be2,
                                                   p_lw, p_lb, src_mask, out + o_pitch);
  // x2 = x + pitch_emb[searchsorted(pitch_bins, pitch_target)]
  add_emb_kernel<<<grow, blk, 0, stream>>>(x, pitch_t, pitch_bins, pitch_emb, buf1);
  // energy predictor (on x2)
  vp_conv_ln_kernel    <<<gconv, blk, 0, stream>>>(buf1, e_w1, e_b1, e_g1, e_be1, buf0);
  vp_conv_ln_lin_kernel<<<gconv, blk, 0, stream>>>(buf0, e_w2, e_b2, e_g2, e_be2,
                                                   e_lw, e_lb, src_mask, out + o_energy);
  // x3 = x2 + energy_emb (in place: each thread reads its own element first)
  add_emb_kernel<<<grow, blk, 0, stream>>>(buf1, energy_t, energy_bins, energy_emb, buf1);
  // cumsum + duration_rounded + mel_len
  cumsum_kernel<<<dim3(B_), dim3(512), 0, stream>>>(dur_t, cum, out + o_dur, out + o_mellen);
  // length regulator -> expanded x + mel_mask
  lenreg_kernel<<<glr, blk, 0, stream>>>(buf1, cum, out + o_x, out + o_melmask);
}